// Prototype_Discriminative_Loss_61959198212231
// MI455X (gfx1250) — compile-verified
//
#include <hip/hip_runtime.h>

// Problem constants (from the reference)
constexpr int NB = 512;   // batch
constexpr int NC = 200;   // classes
constexpr int ND = 768;   // feature dim
constexpr float GAMMA_F = 2.0f;
constexpr float EPS_F = 1e-6f;

constexpr int KCHUNK = 192;            // K staging chunk (divides 768)
constexpr int NTILES = 13;             // ceil(200/16)
constexpr int THREADS = NTILES * 32;   // 416 threads = 13 wave32s

typedef float v2f __attribute__((ext_vector_type(2)));
typedef float v8f __attribute__((ext_vector_type(8)));

// ---------------------------------------------------------------------------
// Kernel 1: per-row argmax of output*candidate (first-max semantics), and
// zero the loss accumulators (block 0).
// ---------------------------------------------------------------------------
__global__ __launch_bounds__(32)
void pdl_argmax_kernel(const float* __restrict__ output,
                       const float* __restrict__ cand,
                       int* __restrict__ idxArr,
                       float* __restrict__ acc) {
    const int b = blockIdx.x;
    const int lane = threadIdx.x;

    if (b == 0 && lane < 3) acc[lane] = 0.0f;

    float best = -3.4028235e38f;
    int bi = NC;
    for (int c = lane; c < NC; c += 32) {
        float v = output[b * NC + c] * cand[b * NC + c];
        if (v > best) { best = v; bi = c; }   // strict > keeps first occurrence
    }
    // cross-lane reduce: larger value wins; ties -> smaller index (first max)
    for (int off = 16; off > 0; off >>= 1) {
        float ov = __shfl_xor(best, off, 32);
        int   oi = __shfl_xor(bi,  off, 32);
        if (ov > best || (ov == best && oi < bi)) { best = ov; bi = oi; }
    }
    if (lane == 0) idxArr[b] = bi;
}

// ---------------------------------------------------------------------------
// Kernel 2: gathered-row fp32 GEMM via V_WMMA_F32_16X16X4_F32 + loss sums.
// Grid: NB/16 blocks of 16 batch rows; 13 waves, one 16-class tile each.
// ---------------------------------------------------------------------------
__global__ __launch_bounds__(THREADS)
void pdl_wmma_kernel(const float* __restrict__ fw,
                     const float* __restrict__ fs,
                     const float* __restrict__ proto,
                     const int* __restrict__ idxArr,
                     float* __restrict__ acc) {
    __shared__ float lds_w[16 * KCHUNK];   // gathered rows, weak branch
    __shared__ float lds_s[16 * KCHUNK];   // gathered rows, strong branch
    __shared__ int   sh_idx[16];

    const int tid  = threadIdx.x;
    const int bb   = blockIdx.x * 16;      // batch tile base
    const int wave = tid >> 5;
    const int lane = tid & 31;
    const int nn   = lane & 15;            // N within tile (B/C/D: N = lane&15)
    const int hi   = lane >> 4;            // lane group selects K pair / M half
    const int n0   = wave * 16;
    const int ng   = n0 + nn;              // global class column
    const bool nvalid = (ng < NC);
    const int ngc  = nvalid ? ng : (NC - 1);   // clamp: no divergence, junk cols discarded

    if (tid < 16) sh_idx[tid] = idxArr[bb + tid];

    v8f cw = {0.f,0.f,0.f,0.f,0.f,0.f,0.f,0.f};
    v8f cs = {0.f,0.f,0.f,0.f,0.f,0.f,0.f,0.f};

    const int row = nn;                    // A-matrix: M = lane&15

    for (int k0 = 0; k0 < ND; k0 += KCHUNK) {
        __syncthreads();                   // protect LDS from previous chunk readers
        // Cooperative stage of the gathered 16xKCHUNK A tiles (both branches)
        for (int t = tid; t < 16 * KCHUNK; t += THREADS) {
            const int r = t / KCHUNK;
            const int d = t - r * KCHUNK;
            const size_t src = ((size_t)(bb + r) * NC + (size_t)sh_idx[r]) * ND
                             + (size_t)(k0 + d);
            lds_w[t] = fw[src];
            lds_s[t] = fs[src];
        }
        __syncthreads();

        for (int k = 0; k < KCHUNK; k += 4) {
            const int kk = k + hi * 2;     // ISA layout: lanes 0-15 -> K{0,1}, 16-31 -> K{2,3}
            v2f aw, as_, bfr;
            aw.x  = lds_w[row * KCHUNK + kk];
            aw.y  = lds_w[row * KCHUNK + kk + 1];
            as_.x = lds_s[row * KCHUNK + kk];
            as_.y = lds_s[row * KCHUNK + kk + 1];
            const size_t p = (size_t)ngc * ND + (size_t)(k0 + kk);
            bfr.x = proto[p];
            bfr.y = proto[p + 1];
            // D = A x B + C, fp32 WMMA (args: neg_a, A, neg_b, B, c_mod, C, reuse_a, reuse_b)
            cw = __builtin_amdgcn_wmma_f32_16x16x4_f32(false, aw,  false, bfr,
                                                       (short)0, cw, false, false);
            cs = __builtin_amdgcn_wmma_f32_16x16x4_f32(false, as_, false, bfr,
                                                       (short)0, cs, false, false);
        }
    }

    // C/D layout: VGPR r, lanes 0-15 -> M=r, lanes 16-31 -> M=r+8; N = lane&15
    float l_pos = 0.f, l_apos = 0.f, l_abs = 0.f;
    if (nvalid) {
        #pragma unroll
        for (int r = 0; r < 8; ++r) {
            const int m = r + hi * 8;
            const float vw = cw[r];
            const float vs = cs[r];
            const float a = fabsf(vw) + fabsf(vs);
            l_abs += a;
            if (ng == sh_idx[m]) { l_pos += vw + vs; l_apos += a; }
        }
    }
    // wave32 reduction, then 3 atomics per wave
    for (int off = 16; off > 0; off >>= 1) {
        l_pos  += __shfl_xor(l_pos,  off, 32);
        l_apos += __shfl_xor(l_apos, off, 32);
        l_abs  += __shfl_xor(l_abs,  off, 32);
    }
    if (lane == 0) {
        atomicAdd(&acc[0], l_pos);
        atomicAdd(&acc[1], l_apos);
        atomicAdd(&acc[2], l_abs);
    }
}

// ---------------------------------------------------------------------------
// Kernel 3: fold the three sums into the scalar loss.
// loss = 2 - S_pos/(B(1+eps)) + GAMMA*(S_abs - S_abspos)/(B(C-1+eps))
// ---------------------------------------------------------------------------
__global__ void pdl_finalize_kernel(const float* __restrict__ acc,
                                    float* __restrict__ out) {
    if (threadIdx.x == 0 && blockIdx.x == 0) {
        const float s_pos  = acc[0];
        const float s_apos = acc[1];
        const float s_abs  = acc[2];
        const float loss = 2.0f
            - s_pos / ((float)NB * (1.0f + EPS_F))
            + GAMMA_F * (s_abs - s_apos) / ((float)NB * ((float)(NC - 1) + EPS_F));
        out[0] = loss;
    }
}

extern "C" void kernel_launch(void* const* d_in, const int* in_sizes, int n_in,
                              void* d_out, int out_size, void* d_ws, size_t ws_size,
                              hipStream_t stream) {
    const float* fw     = (const float*)d_in[0];  // features_w [B,C,D]
    const float* fs     = (const float*)d_in[1];  // features_s [B,C,D]
    const float* proto  = (const float*)d_in[2];  // prototypes [C,D]
    const float* output = (const float*)d_in[3];  // output [B,C]
    const float* cand   = (const float*)d_in[4];  // candidate_label [B,C]

    int*   idxArr = (int*)d_ws;                          // NB ints
    float* acc    = (float*)((char*)d_ws + NB * sizeof(int)); // 3 floats

    pdl_argmax_kernel<<<NB, 32, 0, stream>>>(output, cand, idxArr, acc);
    pdl_wmma_kernel<<<NB / 16, THREADS, 0, stream>>>(fw, fs, proto, idxArr, acc);
    pdl_finalize_kernel<<<1, 32, 0, stream>>>(acc, (float*)d_out);
}